// MPNEncoder_24824910971089
// MI455X (gfx1250) — compile-verified
//
#include <hip/hip_runtime.h>

#define NBOND 500000
#define NATOM 250000
#define MAXNB 6
#define DMSG 165
#define DATT 151
#define HDIM 128

typedef __attribute__((ext_vector_type(2))) float v2f;
typedef __attribute__((ext_vector_type(8))) float v8f;

// ---------------------------------------------------------------------------
// Kernel 1: inp = init_messages @ W_i   (K = 165 -> pad 168), msg0 = relu(inp)
// ---------------------------------------------------------------------------
#define K1_KP  168
#define K1_STR 172   // 172 % 8 == 4 -> conflict-free 16x4 fragment reads
#define K1_WAVES 4

__global__ __launch_bounds__(K1_WAVES * 32)
void k_init_gemm(const float* __restrict__ A, const float* __restrict__ W,
                 float* __restrict__ inp, float* __restrict__ msg0, int nb)
{
    __shared__ float ldsW[HDIM * K1_STR];            // transposed: [n][k]
    __shared__ float ldsA[K1_WAVES * 16 * K1_STR];   // [wave*16+row][k]
    const int tid   = threadIdx.x;
    const int wv    = tid >> 5;
    const int lane  = tid & 31;
    const int upper = lane >> 4;
    const int l15   = lane & 15;
    const int r0    = blockIdx.x * (K1_WAVES * 16) + wv * 16;

    for (int i = tid; i < HDIM * K1_KP; i += blockDim.x) {
        int k = i >> 7, n = i & (HDIM - 1);
        ldsW[n * K1_STR + k] = (k < DMSG) ? W[k * HDIM + n] : 0.0f;
    }
    for (int r = 0; r < 16; ++r) {
        int b = r0 + r;
        for (int c = lane; c < K1_KP; c += 32) {
            float v = 0.0f;
            if (b < nb && c < DMSG) v = A[(size_t)b * DMSG + c];
            ldsA[(wv * 16 + r) * K1_STR + c] = v;
        }
    }
    __syncthreads();

    v8f zero = {};
    v8f acc[8];
    #pragma unroll
    for (int t = 0; t < 8; ++t) acc[t] = zero;

    for (int k = 0; k < K1_KP; k += 4) {
        int k2 = k + 2 * upper;
        v2f af = *(const v2f*)&ldsA[(wv * 16 + l15) * K1_STR + k2];
        #pragma unroll
        for (int t = 0; t < 8; ++t) {
            v2f bf = *(const v2f*)&ldsW[(t * 16 + l15) * K1_STR + k2];
            acc[t] = __builtin_amdgcn_wmma_f32_16x16x4_f32(
                false, af, false, bf, (short)0, acc[t], false, false);
        }
    }

    #pragma unroll
    for (int t = 0; t < 8; ++t) {
        int col = t * 16 + l15;
        #pragma unroll
        for (int j = 0; j < 8; ++j) {
            int r = r0 + j + 8 * upper;
            if (r < nb) {
                float v = acc[t][j];
                size_t o = (size_t)r * HDIM + col;
                inp[o]  = v;
                msg0[o] = v > 0.0f ? v : 0.0f;
            }
        }
    }
}

// ---------------------------------------------------------------------------
// Kernel 2: attached_nei[a] = sum_j attached_fea[a2attached[a][j]]  (D=151)
// ---------------------------------------------------------------------------
__global__ __launch_bounds__(256)
void k_att_gather(const float* __restrict__ att, const int* __restrict__ a2att,
                  float* __restrict__ out, int na)
{
    int wv = threadIdx.x >> 5, lane = threadIdx.x & 31;
    int a = blockIdx.x * 8 + wv;
    if (a >= na) return;
    int idx[MAXNB];
    #pragma unroll
    for (int j = 0; j < MAXNB; ++j) idx[j] = a2att[a * MAXNB + j];
    for (int c = lane; c < DATT; c += 32) {
        float s = 0.0f;
        #pragma unroll
        for (int j = 0; j < MAXNB; ++j) s += att[(size_t)idx[j] * DATT + c];
        out[(size_t)a * DATT + c] = s;
    }
}

// ---------------------------------------------------------------------------
// Kernel 3: C = inp + (att_nei[b2a] - att_fea[b2a[b2revb]]) @ W_h[128:279,:]
//           (K = 151 -> pad 152), in-place update of inp buffer
// ---------------------------------------------------------------------------
#define K3_KP  152
#define K3_STR 156   // 156 % 8 == 4
#define K3_WAVES 4

__global__ __launch_bounds__(K3_WAVES * 32)
void k_c_gemm(const float* __restrict__ attN, const float* __restrict__ attF,
              const float* __restrict__ Wh,
              const int* __restrict__ b2a, const int* __restrict__ b2revb,
              float* __restrict__ inpC, int nb)
{
    __shared__ float ldsW[HDIM * K3_STR];
    __shared__ float ldsA[K3_WAVES * 16 * K3_STR];
    const int tid   = threadIdx.x;
    const int wv    = tid >> 5;
    const int lane  = tid & 31;
    const int upper = lane >> 4;
    const int l15   = lane & 15;
    const int r0    = blockIdx.x * (K3_WAVES * 16) + wv * 16;

    for (int i = tid; i < HDIM * K3_KP; i += blockDim.x) {
        int k = i >> 7, n = i & (HDIM - 1);
        ldsW[n * K3_STR + k] = (k < DATT) ? Wh[(size_t)(HDIM + k) * HDIM + n] : 0.0f;
    }
    for (int r = 0; r < 16; ++r) {
        int b = r0 + r;
        int a = 0, arev = 0;
        bool valid = (b < nb);
        if (valid) {
            a    = b2a[b];
            arev = b2a[b2revb[b]];
        }
        for (int c = lane; c < K3_KP; c += 32) {
            float v = 0.0f;
            if (valid && c < DATT)
                v = attN[(size_t)a * DATT + c] - attF[(size_t)arev * DATT + c];
            ldsA[(wv * 16 + r) * K3_STR + c] = v;
        }
    }
    __syncthreads();

    v8f zero = {};
    v8f acc[8];
    #pragma unroll
    for (int t = 0; t < 8; ++t) acc[t] = zero;

    for (int k = 0; k < K3_KP; k += 4) {
        int k2 = k + 2 * upper;
        v2f af = *(const v2f*)&ldsA[(wv * 16 + l15) * K3_STR + k2];
        #pragma unroll
        for (int t = 0; t < 8; ++t) {
            v2f bf = *(const v2f*)&ldsW[(t * 16 + l15) * K3_STR + k2];
            acc[t] = __builtin_amdgcn_wmma_f32_16x16x4_f32(
                false, af, false, bf, (short)0, acc[t], false, false);
        }
    }

    #pragma unroll
    for (int t = 0; t < 8; ++t) {
        int col = t * 16 + l15;
        #pragma unroll
        for (int j = 0; j < 8; ++j) {
            int r = r0 + j + 8 * upper;
            if (r < nb) {
                size_t o = (size_t)r * HDIM + col;
                inpC[o] = inpC[o] + acc[t][j];
            }
        }
    }
}

// ---------------------------------------------------------------------------
// Kernel 4a: nei[a] = sum_j message[a2nei[a][j]]  (H = 128, float4)
// ---------------------------------------------------------------------------
__global__ __launch_bounds__(256)
void k_nei_gather(const float* __restrict__ msg, const int* __restrict__ a2nei,
                  float* __restrict__ out, int na)
{
    int wv = threadIdx.x >> 5, lane = threadIdx.x & 31;
    int a = blockIdx.x * 8 + wv;
    if (a >= na) return;
    int idx[MAXNB];
    #pragma unroll
    for (int j = 0; j < MAXNB; ++j) idx[j] = a2nei[a * MAXNB + j];
    float4 s = make_float4(0.f, 0.f, 0.f, 0.f);
    #pragma unroll
    for (int j = 0; j < MAXNB; ++j) {
        const float4 v = *(const float4*)(msg + (size_t)idx[j] * HDIM + lane * 4);
        s.x += v.x; s.y += v.y; s.z += v.z; s.w += v.w;
    }
    *(float4*)(out + (size_t)a * HDIM + lane * 4) = s;
}

// ---------------------------------------------------------------------------
// Kernel 4b: msg_out = relu(C + (nei[b2a] - msg_in[b2revb]) @ W_h[0:128,:])
// ---------------------------------------------------------------------------
#define KB_STR 132   // 132 % 8 == 4
#define KB_WAVES 8

__global__ __launch_bounds__(KB_WAVES * 32)
void k_bond_gemm(const float* __restrict__ nei, const float* __restrict__ msg_in,
                 const float* __restrict__ Cbuf, const float* __restrict__ Wh,
                 const int* __restrict__ b2a, const int* __restrict__ b2revb,
                 float* __restrict__ msg_out, int nb)
{
    __shared__ float ldsW[HDIM * KB_STR];
    __shared__ float ldsA[KB_WAVES * 16 * KB_STR];
    const int tid   = threadIdx.x;
    const int wv    = tid >> 5;
    const int lane  = tid & 31;
    const int upper = lane >> 4;
    const int l15   = lane & 15;
    const int r0    = blockIdx.x * (KB_WAVES * 16) + wv * 16;

    for (int i = tid; i < HDIM * HDIM; i += blockDim.x) {
        int k = i >> 7, n = i & (HDIM - 1);
        ldsW[n * KB_STR + k] = Wh[i];            // Wh row-major: i = k*128 + n
    }
    for (int r = 0; r < 16; ++r) {
        int b = r0 + r;
        float4 v = make_float4(0.f, 0.f, 0.f, 0.f);
        if (b < nb) {
            int a  = b2a[b];
            int rb = b2revb[b];
            const float4 x = *(const float4*)(nei    + (size_t)a  * HDIM + lane * 4);
            const float4 y = *(const float4*)(msg_in + (size_t)rb * HDIM + lane * 4);
            v.x = x.x - y.x; v.y = x.y - y.y; v.z = x.z - y.z; v.w = x.w - y.w;
        }
        *(float4*)&ldsA[(wv * 16 + r) * KB_STR + lane * 4] = v;
    }
    __syncthreads();

    v8f zero = {};
    v8f acc[8];
    #pragma unroll
    for (int t = 0; t < 8; ++t) acc[t] = zero;

    #pragma unroll 4
    for (int k = 0; k < HDIM; k += 4) {
        int k2 = k + 2 * upper;
        v2f af = *(const v2f*)&ldsA[(wv * 16 + l15) * KB_STR + k2];
        #pragma unroll
        for (int t = 0; t < 8; ++t) {
            v2f bf = *(const v2f*)&ldsW[(t * 16 + l15) * KB_STR + k2];
            acc[t] = __builtin_amdgcn_wmma_f32_16x16x4_f32(
                false, af, false, bf, (short)0, acc[t], false, false);
        }
    }

    #pragma unroll
    for (int t = 0; t < 8; ++t) {
        int col = t * 16 + l15;
        #pragma unroll
        for (int j = 0; j < 8; ++j) {
            int r = r0 + j + 8 * upper;
            if (r < nb) {
                size_t o = (size_t)r * HDIM + col;
                float v = Cbuf[o] + acc[t][j];
                msg_out[o] = v > 0.0f ? v : 0.0f;
            }
        }
    }
}

// ---------------------------------------------------------------------------
// Host launcher
// ---------------------------------------------------------------------------
extern "C" void kernel_launch(void* const* d_in, const int* in_sizes, int n_in,
                              void* d_out, int out_size, void* d_ws, size_t ws_size,
                              hipStream_t stream) {
    const float* init_messages = (const float*)d_in[0];
    const float* att_fea       = (const float*)d_in[1];
    const float* W_i           = (const float*)d_in[2];
    const float* W_h           = (const float*)d_in[3];
    const int*   a2nei         = (const int*)d_in[4];
    const int*   a2att         = (const int*)d_in[5];
    const int*   b2a           = (const int*)d_in[6];
    const int*   b2revb        = (const int*)d_in[7];
    float*       out           = (float*)d_out;

    float* ws = (float*)d_ws;
    size_t off = 0;
    float* inpC = ws + off; off += (size_t)NBOND * HDIM;   // inp, then C (in place)
    float* msgA = ws + off; off += (size_t)NBOND * HDIM;
    float* msgB = ws + off; off += (size_t)NBOND * HDIM;
    float* neiB = ws + off; off += (size_t)NATOM * HDIM;
    float* attN = ws + off; off += (size_t)NATOM * DATT;

    const int gInit = (NBOND + K1_WAVES * 16 - 1) / (K1_WAVES * 16);
    const int gC    = (NBOND + K3_WAVES * 16 - 1) / (K3_WAVES * 16);
    const int gBond = (NBOND + KB_WAVES * 16 - 1) / (KB_WAVES * 16);
    const int gAtom = (NATOM + 7) / 8;

    // One-time: input GEMM + relu, constant gather, constant residual C
    k_init_gemm<<<gInit, K1_WAVES * 32, 0, stream>>>(init_messages, W_i, inpC, msgA, NBOND);
    k_att_gather<<<gAtom, 256, 0, stream>>>(att_fea, a2att, attN, NATOM);
    k_c_gemm<<<gC, K3_WAVES * 32, 0, stream>>>(attN, att_fea, W_h, b2a, b2revb, inpC, NBOND);

    // 5 message-passing iterations; last writes directly to d_out
    float* cur = msgA;
    float* nxt = msgB;
    for (int it = 0; it < 5; ++it) {
        k_nei_gather<<<gAtom, 256, 0, stream>>>(cur, a2nei, neiB, NATOM);
        float* dst = (it == 4) ? out : nxt;
        k_bond_gemm<<<gBond, KB_WAVES * 32, 0, stream>>>(neiB, cur, inpC, W_h,
                                                         b2a, b2revb, dst, NBOND);
        float* tmp = cur; cur = dst; nxt = tmp;
    }
}